// EikonalLoss_41180146434707
// MI455X (gfx1250) — compile-verified
//
#include <hip/hip_runtime.h>
#include <stdint.h>

// ---------------------------------------------------------------------------
// EikonalLoss: 3x3 Sobel (edge-padded) -> |sqrt(gx^2+gy^2+eps)-1| masked mean.
// Bandwidth-bound (128 MiB in, 4 B out; ~5.7us floor @ 23.3 TB/s).
// CDNA5 path: TDM tensor_load_to_lds DMA for the pred row band, overlapped
// with float4 streaming of the reachable mask; LDS stencil; f64 atomics.
// ---------------------------------------------------------------------------

typedef __attribute__((ext_vector_type(4))) unsigned int tdm_g0_t;
typedef __attribute__((ext_vector_type(8))) int          tdm_g1_t;
typedef __attribute__((ext_vector_type(4))) int          tdm_g23_t;

constexpr int kW       = 512;
constexpr int kH       = 512;
constexpr int kTileH   = 16;
constexpr int kThreads = 256;
constexpr int kPasses  = (kTileH * kW) / (kThreads * 4);  // 8

__global__ __launch_bounds__(kThreads)
void eik_main_kernel(const float* __restrict__ pred,
                     const float* __restrict__ reach,
                     double* __restrict__ ws) {
  __shared__ float smem[(kTileH + 2) * kW];  // 18 x 512 f32 = 36 KB

  const int tid         = threadIdx.x;
  const int bandsPerImg = kH / kTileH;
  const int img         = blockIdx.x / bandsPerImg;
  const int band        = blockIdx.x % bandsPerImg;
  const int y0          = band * kTileH;
  const int ylo         = (y0 - 1 < 0) ? 0 : (y0 - 1);
  const int yhi         = (y0 + kTileH > kH - 1) ? (kH - 1) : (y0 + kTileH);
  const int nrows       = yhi - ylo + 1;  // 17 at image edges, else 18

  // --- TDM bulk DMA: rows [ylo..yhi] x 512 f32 -> LDS (wave 0 issues once) ---
  if (tid < 32) {
    uint64_t gaddr = (uint64_t)(uintptr_t)(pred + (size_t)img * (kH * kW)
                                                + (size_t)ylo * kW);
    uint32_t laddr = (uint32_t)(uintptr_t)(&smem[0]);  // flat LDS addr low bits

    // D# group 0 (ISA 8.3): count=1(valid), lds_addr, global_addr[56:0], type=2
    tdm_g0_t g0;
    g0.x = 1u;
    g0.y = laddr;
    g0.z = (uint32_t)gaddr;
    g0.w = ((uint32_t)(gaddr >> 32) & 0x01FFFFFFu) | 0x80000000u;

    // D# group 1 (ISA 8.4): data_size=4B; tensor==tile (no OOB), row stride 512
    tdm_g1_t g1;
    g1.s0 = 0x00020000;                    // wg_mask=0, data_size=2 (4B)
    g1.s1 = (int)((unsigned)kW << 16);     // tensor_dim0 lo16 @bits[63:48]
    g1.s2 = (int)((unsigned)nrows << 16);  // tensor_dim0 hi16=0 | tensor_dim1 lo16
    g1.s3 = (int)((unsigned)kW << 16);     // tensor_dim1 hi16=0 | tile_dim0=512
    g1.s4 = nrows;                         // tile_dim1=nrows, tile_dim2=0
    g1.s5 = kW;                            // tensor_dim0_stride lo32 (elements)
    g1.s6 = 0;
    g1.s7 = 0;

    tdm_g23_t gz  = {0, 0, 0, 0};          // 2D tensor: groups 2/3 unused
    tdm_g1_t  gz8 = {0, 0, 0, 0, 0, 0, 0, 0};

    __builtin_amdgcn_tensor_load_to_lds(g0, g1, gz, gz, gz8, 0);
  }

  // --- overlap: stream reachable (float4) while the DMA is in flight --------
  const float4* r4 = reinterpret_cast<const float4*>(reach + (size_t)img * (kH * kW));
  uint32_t mbits = 0;
  #pragma unroll
  for (int j = 0; j < kPasses; ++j) {
    const int p  = j * (kThreads * 4) + tid * 4;
    const int y  = y0 + (p >> 9);
    const int x4 = p & (kW - 1);
    const float4 rv = r4[(y * kW + x4) >> 2];
    mbits |= (rv.x > 0.5f ? 1u : 0u) << (4 * j + 0);
    mbits |= (rv.y > 0.5f ? 1u : 0u) << (4 * j + 1);
    mbits |= (rv.z > 0.5f ? 1u : 0u) << (4 * j + 2);
    mbits |= (rv.w > 0.5f ? 1u : 0u) << (4 * j + 3);
  }
  const float msum = (float)__popc(mbits);

  __builtin_amdgcn_s_wait_tensorcnt(0);  // no-op for waves with TENSORcnt==0
  __syncthreads();

  // --- stencil: 4 adjacent pixels per thread, 3x6 LDS row segments ---------
  float vsum = 0.f;
  #pragma unroll
  for (int j = 0; j < kPasses; ++j) {
    const int p  = j * (kThreads * 4) + tid * 4;
    const int y  = y0 + (p >> 9);
    const int x4 = p & (kW - 1);
    const int rm = ((((y - 1) < ylo) ? ylo : (y - 1)) - ylo) * kW;
    const int r0 = (y - ylo) * kW;
    const int rp = ((((y + 1) > yhi) ? yhi : (y + 1)) - ylo) * kW;

    float U[6], M[6], B[6];
    #pragma unroll
    for (int c = 0; c < 6; ++c) {
      int col = x4 - 1 + c;
      col = (col < 0) ? 0 : ((col > kW - 1) ? (kW - 1) : col);
      U[c] = smem[rm + col];
      M[c] = smem[r0 + col];
      B[c] = smem[rp + col];
    }
    #pragma unroll
    for (int k = 0; k < 4; ++k) {
      const float a = U[k], b = U[k + 1], c = U[k + 2];
      const float d = M[k],               e = M[k + 2];
      const float f = B[k], g = B[k + 1], h = B[k + 2];
      // correlation (no flip), kernels already /8
      const float gx = ((c - a) + 2.f * (e - d) + (h - f)) * 0.125f;
      const float gy = ((f - a) + 2.f * (g - b) + (h - c)) * 0.125f;
      const float mag  = sqrtf(fmaf(gx, gx, fmaf(gy, gy, 1e-8f)));
      const float viol = fabsf(mag - 1.f);
      const float m    = ((mbits >> (4 * j + k)) & 1u) ? 1.f : 0.f;
      vsum = fmaf(viol, m, vsum);
    }
  }

  // --- block reduction (reuse smem after all stencil reads are done) -------
  __syncthreads();
  smem[tid]            = vsum;
  smem[kThreads + tid] = msum;
  __syncthreads();
  for (int s = kThreads / 2; s > 0; s >>= 1) {
    if (tid < s) {
      smem[tid]            += smem[tid + s];
      smem[kThreads + tid] += smem[kThreads + tid + s];
    }
    __syncthreads();
  }
  if (tid == 0) {
    atomicAdd(&ws[0], (double)smem[0]);          // global_atomic_add_f64
    atomicAdd(&ws[1], (double)smem[kThreads]);
  }
}

__global__ void eik_init_kernel(double* ws) {
  if (threadIdx.x < 2) ws[threadIdx.x] = 0.0;
}

__global__ void eik_final_kernel(const double* __restrict__ ws,
                                 float* __restrict__ out) {
  const double denom = (ws[1] < 1.0) ? 1.0 : ws[1];
  out[0] = (float)(ws[0] / denom);
}

extern "C" void kernel_launch(void* const* d_in, const int* in_sizes, int n_in,
                              void* d_out, int out_size, void* d_ws, size_t ws_size,
                              hipStream_t stream) {
  const float* pred  = (const float*)d_in[0];
  const float* reach = (const float*)d_in[1];
  float*  out = (float*)d_out;
  double* ws  = (double*)d_ws;

  const int nimg   = in_sizes[0] / (kH * kW);        // 64
  const int blocks = nimg * (kH / kTileH);           // 2048

  eik_init_kernel<<<1, 32, 0, stream>>>(ws);
  eik_main_kernel<<<blocks, kThreads, 0, stream>>>(pred, reach, ws);
  eik_final_kernel<<<1, 1, 0, stream>>>(ws, out);
}